// MyModel_87454124082256
// MI455X (gfx1250) — compile-verified
//
#include <hip/hip_runtime.h>
#include <stdint.h>

// Problem constants (from the reference):
//   inputs [2,256,256,1], ind1 [367,217,721,2], w1 [367,217,721,1,1], lambda scalar
#define K_TOT   367
#define P_DIM   217
#define Q_DIM   721
#define PQ      (P_DIM * Q_DIM)       // 156457 outputs per batch
#define NTHR    256
#define NWAVE   (NTHR / 32)
#define KC      8                     // k-chunk staged per LDS buffer
#define NCHUNK  ((K_TOT + KC - 1) / KC)   // 46

// Cache policy for the streamed (never-reused-in-pass) operands: TH_LOAD_NT.
// ind1+w1 are 689 MB streamed 5x (>> 192 MB L2); NT keeps them from evicting
// the small y field we gather from (<= 1.25 MB, L2-resident).
#define CPOL_NT 1

// Pointer types matching the async-to-LDS builtin prototypes:
//   b64: (v2i AS1*, v2i AS3*, imm offset, imm cpol)
//   b32: (int AS1*, int AS3*, imm offset, imm cpol)
typedef int v2i __attribute__((ext_vector_type(2)));
typedef __attribute__((address_space(1))) v2i* g_v2i_p;
typedef __attribute__((address_space(3))) v2i* l_v2i_p;
typedef __attribute__((address_space(1))) int* g_i32_p;
typedef __attribute__((address_space(3))) int* l_i32_p;

// Generic->AS1 is numerically identity; generic->LDS: low 32 bits of a generic
// pointer into __shared__ ARE the LDS byte address.
#define ASYNC_B64(g, l)                                                   \
  __builtin_amdgcn_global_load_async_to_lds_b64(                          \
      (g_v2i_p)(uintptr_t)(g), (l_v2i_p)(uint32_t)(uintptr_t)(l), 0, CPOL_NT)
#define ASYNC_B32(g, l)                                                   \
  __builtin_amdgcn_global_load_async_to_lds_b32(                          \
      (g_i32_p)(uintptr_t)(g), (l_i32_p)(uint32_t)(uintptr_t)(l), 0, CPOL_NT)

template <int N>
static __device__ __forceinline__ void wait_asynccnt() {
  // Compiler barrier too: keeps LDS reads below the wait.
  asm volatile("s_wait_asynccnt %0" ::"i"(N) : "memory");
}

// One Jacobi-style pass:
//   ydst[b, n] = lam * sum_k w[k, n] * ysrc[b*srcHW + i0[k,n]*srcW + i1[k,n]]
// for n = p*Q+q in [0, PQ), b in {0,1}.
__global__ __launch_bounds__(NTHR) void gather_iter_kernel(
    const float* __restrict__ ysrc,   // [2, srcH, srcW]
    const int*   __restrict__ ind,    // [K, PQ, 2]
    const float* __restrict__ w,      // [K, PQ]
    const float* __restrict__ lam,    // scalar
    float*       __restrict__ ydst,   // [2, PQ]
    int srcW, int srcHW)
{
  // Per-wave double-buffered staging: no cross-wave barriers needed.
  __shared__ int2  s_ind[2][NWAVE][KC][32];   // 32 KB
  __shared__ float s_w  [2][NWAVE][KC][32];   // 16 KB

  const int tid   = threadIdx.x;
  const int wid   = tid >> 5;
  const int lane  = tid & 31;
  const int n_raw = blockIdx.x * NTHR + tid;
  const bool ok   = (n_raw < PQ);
  const int n     = ok ? n_raw : (PQ - 1);   // clamp: tail lanes compute garbage, never store

  const int2* __restrict__ indv = (const int2*)ind;   // [K][PQ]

  auto stage = [&](int buf, int c) {
    const int k0 = c * KC;
#pragma unroll
    for (int kk = 0; kk < KC; ++kk) {
      int k = k0 + kk;
      if (k >= K_TOT) k = K_TOT - 1;          // tail: harmless duplicate fetch
      ASYNC_B64(&indv[(size_t)k * PQ + n], &s_ind[buf][wid][kk][lane]);
      ASYNC_B32(&w   [(size_t)k * PQ + n], &s_w  [buf][wid][kk][lane]);
    }
  };

  float acc0 = 0.0f, acc1 = 0.0f;

  stage(0, 0);
  for (int c = 0; c < NCHUNK; ++c) {
    const int buf = c & 1;
    if (c + 1 < NCHUNK) {
      stage(buf ^ 1, c + 1);
      // 2*KC ops in flight for chunk c+1; loads complete in order, so
      // ASYNCcnt <= 2*KC guarantees chunk c has fully landed in LDS.
      wait_asynccnt<2 * KC>();
    } else {
      wait_asynccnt<0>();
    }

    const int kmax = (c == NCHUNK - 1) ? (K_TOT - c * KC) : KC;
#pragma unroll
    for (int kk = 0; kk < KC; ++kk) {
      if (kk >= kmax) break;
      const int2  ij = s_ind[buf][wid][kk][lane];
      const float wv = s_w [buf][wid][kk][lane];
      const int off  = ij.x * srcW + ij.y;      // gather hits L2 (y <= 1.25 MB)
      acc0 = fmaf(wv, ysrc[off],         acc0);
      acc1 = fmaf(wv, ysrc[srcHW + off], acc1);
    }
  }

  if (ok) {
    const float lv = lam[0];
    ydst[n]              = lv * acc0;
    ydst[(size_t)PQ + n] = lv * acc1;
  }
}

extern "C" void kernel_launch(void* const* d_in, const int* in_sizes, int n_in,
                              void* d_out, int out_size, void* d_ws, size_t ws_size,
                              hipStream_t stream)
{
  (void)in_sizes; (void)n_in; (void)out_size; (void)ws_size;

  const float* y0  = (const float*)d_in[0];  // [2,256,256,1]
  const int*   ind = (const int*)  d_in[1];  // [367,217,721,2]
  const float* w   = (const float*)d_in[2];  // [367,217,721,1,1]
  const float* lam = (const float*)d_in[3];  // scalar
  float* out = (float*)d_out;                // [2,217,721,1]

  float* ws0 = (float*)d_ws;                 // [2, PQ]
  float* ws1 = ws0 + (size_t)2 * PQ;         // [2, PQ]  (total 2.5 MB of d_ws)

  const int grid = (PQ + NTHR - 1) / NTHR;   // 612 blocks, 8 waves each

  // 5 sequentially dependent passes; iter 1 reads the [2,256,256] input,
  // then ping-pong through workspace, final pass writes d_out directly.
  gather_iter_kernel<<<grid, NTHR, 0, stream>>>(y0,  ind, w, lam, ws0, 256,   256 * 256);
  gather_iter_kernel<<<grid, NTHR, 0, stream>>>(ws0, ind, w, lam, ws1, Q_DIM, PQ);
  gather_iter_kernel<<<grid, NTHR, 0, stream>>>(ws1, ind, w, lam, ws0, Q_DIM, PQ);
  gather_iter_kernel<<<grid, NTHR, 0, stream>>>(ws0, ind, w, lam, ws1, Q_DIM, PQ);
  gather_iter_kernel<<<grid, NTHR, 0, stream>>>(ws1, ind, w, lam, out, Q_DIM, PQ);
}